// CensusConsistencyLayerVectorized_26147760898488
// MI455X (gfx1250) — compile-verified
//
#include <hip/hip_runtime.h>

#define EPSILON 1e-6f
#define A_BINS 256
#define BLOCK 256
#define WAVES_PER_BLOCK (BLOCK / 32)   // wave32 on gfx1250
#define BLOCKS_PER_BATCH 128

typedef float vf4 __attribute__((ext_vector_type(4)));
typedef int   vi4 __attribute__((ext_vector_type(4)));

#if __has_builtin(__builtin_amdgcn_global_load_async_to_lds_b32) && \
    __has_builtin(__builtin_amdgcn_s_wait_asynccnt)
#define HAVE_ASYNC_LDS 1
#else
#define HAVE_ASYNC_LDS 0
#endif

#if HAVE_ASYNC_LDS
typedef __attribute__((address_space(1))) int gint_t;  // global AS
typedef __attribute__((address_space(3))) int lint_t;  // LDS AS
__device__ __forceinline__ void async_lds_b32(const float* gsrc, float* ldst) {
    __builtin_amdgcn_global_load_async_to_lds_b32(
        (gint_t*)(float*)gsrc, (lint_t*)ldst, 0, 0);
}
#endif

// ---------------- Phase 0: zero the segment-sum scratch ----------------
__global__ void k_zero(float* __restrict__ S, int n) {
    int i = blockIdx.x * blockDim.x + threadIdx.x;
    if (i < n) S[i] = 0.0f;
}

// ---------------- Phase 1: per-batch 256-bin segment sum ----------------
// Per-wave privatized LDS histograms (ds_add_f32), then one global
// fp32 atomic per bin per block. Loads use default RT temporal hints so
// P/ids stay resident in the 192MB L2 for phase 2.
__global__ void __launch_bounds__(BLOCK)
k_hist(const float* __restrict__ P, const int* __restrict__ ids,
       float* __restrict__ S, int N) {
    __shared__ float h[WAVES_PER_BLOCK][A_BINS];
    const int b = blockIdx.y;
    for (int i = threadIdx.x; i < WAVES_PER_BLOCK * A_BINS; i += BLOCK)
        (&h[0][0])[i] = 0.0f;
    __syncthreads();

    const int wave = threadIdx.x >> 5;              // wave32
    const vf4* __restrict__ P4 = (const vf4*)(P + (size_t)b * N);
    const vi4* __restrict__ I4 = (const vi4*)(ids + (size_t)b * N);
    const int nvec = N >> 2;

    for (int v = blockIdx.x * BLOCK + threadIdx.x; v < nvec;
         v += gridDim.x * BLOCK) {
        vf4 p  = P4[v];          // global_load_b128 (RT: keep in L2)
        vi4 id = I4[v];          // global_load_b128
        if (id.x >= 0) unsafeAtomicAdd(&h[wave][id.x], p.x);  // ds_add_f32
        if (id.y >= 0) unsafeAtomicAdd(&h[wave][id.y], p.y);
        if (id.z >= 0) unsafeAtomicAdd(&h[wave][id.z], p.z);
        if (id.w >= 0) unsafeAtomicAdd(&h[wave][id.w], p.w);
    }
    // scalar tail (N not multiple of 4) handled by block 0 of each batch
    if (blockIdx.x == 0) {
        const float* Pb = P + (size_t)b * N;
        const int*   Ib = ids + (size_t)b * N;
        for (int i = (nvec << 2) + threadIdx.x; i < N; i += BLOCK) {
            int id = Ib[i];
            if (id >= 0) unsafeAtomicAdd(&h[wave][id], Pb[i]);
        }
    }
    __syncthreads();

    // reduce the per-wave copies: one bin per thread
    float s = 0.0f;
#pragma unroll
    for (int w = 0; w < WAVES_PER_BLOCK; ++w) s += h[w][threadIdx.x];
    unsafeAtomicAdd(&S[b * A_BINS + threadIdx.x], s);  // global_atomic_add_f32
}

// ---- Phase 2 (fused ratio+apply): out = valid ? p * C[id]/(S[id]+eps) : p ----
__global__ void __launch_bounds__(BLOCK)
k_apply(const float* __restrict__ P, const int* __restrict__ ids,
        const float* __restrict__ S, const float* __restrict__ census,
        float* __restrict__ out, int N) {
    __shared__ float sR[A_BINS];   // becomes the ratio table
    __shared__ float sC[A_BINS];
    const int b = blockIdx.y;
    const int t = threadIdx.x;     // BLOCK == A_BINS

    // Stage this batch's raw-sum row and census row (1KB each) into LDS
    // via gfx1250 async global->LDS DMA, then fold into the ratio table.
#if HAVE_ASYNC_LDS
    async_lds_b32(S + (size_t)b * A_BINS + t,      &sR[t]);
    async_lds_b32(census + (size_t)b * A_BINS + t, &sC[t]);
    __builtin_amdgcn_s_wait_asynccnt(0);
#else
    sR[t] = S[(size_t)b * A_BINS + t];
    sC[t] = census[(size_t)b * A_BINS + t];
#endif
    __syncthreads();
    sR[t] = sC[t] / (sR[t] + EPSILON);
    __syncthreads();

    const vf4* __restrict__ P4 = (const vf4*)(P + (size_t)b * N);
    const vi4* __restrict__ I4 = (const vi4*)(ids + (size_t)b * N);
    vf4*       __restrict__ O4 = (vf4*)(out + (size_t)b * N);
    const int nvec = N >> 2;

    for (int v = blockIdx.x * BLOCK + threadIdx.x; v < nvec;
         v += gridDim.x * BLOCK) {
        // Last-use loads: non-temporal so they free L2 as we stream.
        vf4 p  = __builtin_nontemporal_load(&P4[v]);
        vi4 id = __builtin_nontemporal_load(&I4[v]);
        vf4 o;
        o.x = (id.x >= 0) ? p.x * sR[id.x] : p.x;   // ds_load gather
        o.y = (id.y >= 0) ? p.y * sR[id.y] : p.y;
        o.z = (id.z >= 0) ? p.z * sR[id.z] : p.z;
        o.w = (id.w >= 0) ? p.w * sR[id.w] : p.w;
        // Streaming store: don't evict P/ids lines still needed in L2.
        __builtin_nontemporal_store(o, &O4[v]);
    }
    if (blockIdx.x == 0) {
        const float* Pb = P + (size_t)b * N;
        const int*   Ib = ids + (size_t)b * N;
        float*       Ob = out + (size_t)b * N;
        for (int i = (nvec << 2) + threadIdx.x; i < N; i += BLOCK) {
            int id = Ib[i];
            float p = Pb[i];
            Ob[i] = (id >= 0) ? p * sR[id] : p;
        }
    }
}

extern "C" void kernel_launch(void* const* d_in, const int* in_sizes, int n_in,
                              void* d_out, int out_size, void* d_ws, size_t ws_size,
                              hipStream_t stream) {
    const float* P      = (const float*)d_in[0];   // (B,1,H,W) f32
    const int*   ids    = (const int*)d_in[1];     // (B,H,W)  i32
    const float* census = (const float*)d_in[2];   // (B,A)    f32
    float*       out    = (float*)d_out;

    const int censusN = in_sizes[2];          // B * 256
    const int B       = censusN / A_BINS;     // A is MAX_ADMIN = 256
    const int N       = in_sizes[0] / B;      // pixels per batch (H*W)

    float* S = (float*)d_ws;                  // B*A floats of raw segment sums

    k_zero<<<(censusN + BLOCK - 1) / BLOCK, BLOCK, 0, stream>>>(S, censusN);

    dim3 grid(BLOCKS_PER_BATCH, B);
    k_hist<<<grid, BLOCK, 0, stream>>>(P, ids, S, N);
    k_apply<<<grid, BLOCK, 0, stream>>>(P, ids, S, census, out, N);
}